// MultiResHashEncoding_45226005627123
// MI455X (gfx1250) — compile-verified
//
#include <hip/hip_runtime.h>
#include <math.h>

#define N_LEVELS 16
#define TABLE_SIZE (1u << 19)
#define TMASK (TABLE_SIZE - 1u)
#define PRIME_Y 2654435761u
#define PRIME_Z 805459861u
#define PTS_PER_BLOCK 16
#define BLOCK_THREADS 256 /* 16 points x 16 levels = 8 wave32s */

typedef float f2 __attribute__((ext_vector_type(2)));

typedef __attribute__((address_space(1))) int gas_int; // global
typedef __attribute__((address_space(3))) int las_int; // LDS

struct ResList { int r[N_LEVELS]; };

#if defined(__has_builtin)
#if __has_builtin(__builtin_amdgcn_global_load_async_to_lds_b32)
#define HAVE_ASYNC_LDS 1
#endif
#if __has_builtin(__builtin_amdgcn_s_wait_asynccnt)
#define HAVE_WAIT_ASYNC 1
#endif
#endif

__global__ __launch_bounds__(BLOCK_THREADS)
void hashgrid_encode_kernel(const float* __restrict__ x,
                            const float* __restrict__ tables,
                            float* __restrict__ out,
                            int B, ResList res)
{
    __shared__ float sx[PTS_PER_BLOCK * 3];
    __shared__ int   sres[N_LEVELS];

    const int t = threadIdx.x;
    const long long b0     = (long long)blockIdx.x * PTS_PER_BLOCK;
    const long long xbase  = b0 * 3;
    const long long xtotal = (long long)B * 3;
    long long rem = xtotal - xbase;
    const int navail = (rem < (long long)(PTS_PER_BLOCK * 3)) ? (int)rem
                                                              : (PTS_PER_BLOCK * 3);

    if (t < N_LEVELS) sres[t] = res.r[t];

    // ---- Stage the 16 points' coords into LDS via the CDNA5 async copy path ----
    if (t < navail) {
        const float* gp = x + xbase + t;
#if defined(HAVE_ASYNC_LDS)
        gas_int* gap = (gas_int*)(unsigned long long)gp;
        las_int* lap = (las_int*)(unsigned)(unsigned long long)(&sx[t]);
        __builtin_amdgcn_global_load_async_to_lds_b32(gap, lap, 0, 0);
#else
        sx[t] = *gp;
#endif
    }
#if defined(HAVE_ASYNC_LDS)
#if defined(HAVE_WAIT_ASYNC)
    __builtin_amdgcn_s_wait_asynccnt(0);
#else
    asm volatile("s_wait_asynccnt 0" ::: "memory");
#endif
#endif
    __syncthreads();

    // Warm L2 with upcoming coordinate data (global_prefetch_b8, speculative).
    if (t == 0) {
        long long nb = xbase + (long long)PTS_PER_BLOCK * 3 * 4;
        if (nb < xtotal) __builtin_prefetch(x + nb, 0, 1);
    }

    const int  l    = t & (N_LEVELS - 1);   // level minor -> coalesced stores
    const int  pidx = t >> 4;               // point within block
    const long long b = b0 + pidx;
    if (b >= (long long)B) return;

    const float px = sx[pidx * 3 + 0];
    const float py = sx[pidx * 3 + 1];
    const float pz = sx[pidx * 3 + 2];

    const float r  = (float)sres[l];
    const float xs = px * r, ys = py * r, zs = pz * r;

    const float flx = floorf(xs), fly = floorf(ys), flz = floorf(zs);
    const float frx = xs - flx,   fry = ys - fly,   frz = zs - flz;

    // coords are non-negative (x in [0,1)); only low 19 bits of the int64
    // prime products survive the % 2^19, so u32 arithmetic is exact.
    const unsigned ix0 = (unsigned)(int)flx;
    const unsigned iy0 = (unsigned)(int)fly;
    const unsigned iz0 = (unsigned)(int)flz;
    const unsigned ix1 = (unsigned)(int)ceilf(xs);
    const unsigned iy1 = (unsigned)(int)ceilf(ys);
    const unsigned iz1 = (unsigned)(int)ceilf(zs);

    const unsigned hx0 = ix0;               // prime 1
    const unsigned hx1 = ix1;
    const unsigned hy0 = iy0 * PRIME_Y, hy1 = iy1 * PRIME_Y;
    const unsigned hz0 = iz0 * PRIME_Z, hz1 = iz1 * PRIME_Z;

    unsigned h[8];
    h[0] = (hx0 ^ hy0 ^ hz0) & TMASK;       // bit2=x, bit1=y, bit0=z
    h[1] = (hx0 ^ hy0 ^ hz1) & TMASK;
    h[2] = (hx0 ^ hy1 ^ hz0) & TMASK;
    h[3] = (hx0 ^ hy1 ^ hz1) & TMASK;
    h[4] = (hx1 ^ hy0 ^ hz0) & TMASK;
    h[5] = (hx1 ^ hy0 ^ hz1) & TMASK;
    h[6] = (hx1 ^ hy1 ^ hz0) & TMASK;
    h[7] = (hx1 ^ hy1 ^ hz1) & TMASK;

    const f2* tbl = (const f2*)tables + (size_t)l * TABLE_SIZE;

    // Issue all 8 gathers before consuming any (8 outstanding global_load_b64).
    f2 e[8];
#pragma unroll
    for (int j = 0; j < 8; ++j) e[j] = tbl[h[j]];

    const float gx = 1.0f - frx, gy = 1.0f - fry, gz = 1.0f - frz;
    float w[8];
    w[0] = gx * gy * gz;   w[1] = gx * gy * frz;
    w[2] = gx * fry * gz;  w[3] = gx * fry * frz;
    w[4] = frx * gy * gz;  w[5] = frx * gy * frz;
    w[6] = frx * fry * gz; w[7] = frx * fry * frz;

    float o0 = 0.0f, o1 = 0.0f;
#pragma unroll
    for (int j = 0; j < 8; ++j) {
        o0 = fmaf(w[j], e[j].x, o0);
        o1 = fmaf(w[j], e[j].y, o1);
    }

    // Streaming output: non-temporal b64 store keeps the hash tables hot in L2.
    f2 o; o.x = o0; o.y = o1;
    f2* op = (f2*)(out + (size_t)b * (N_LEVELS * 2) + (l << 1));
    __builtin_nontemporal_store(o, op);
}

extern "C" void kernel_launch(void* const* d_in, const int* in_sizes, int n_in,
                              void* d_out, int out_size, void* d_ws, size_t ws_size,
                              hipStream_t stream) {
    const float* x      = (const float*)d_in[0];
    const float* tables = (const float*)d_in[1];
    float*       out    = (float*)d_out;
    const int B = in_sizes[0] / 3;

    // Replicate numpy's float64 resolution computation exactly:
    //   _b = exp((log(2048) - log(16)) / 15);  res_i = int(16 * _b**i)
    // (level 7 truncates to 153; level 15 is the only boundary-sensitive one)
    ResList rl;
    const double bb = exp((log(2048.0) - log(16.0)) / 15.0);
    for (int i = 0; i < N_LEVELS; ++i)
        rl.r[i] = (int)(16.0 * pow(bb, (double)i));

    const int nblocks = (B + PTS_PER_BLOCK - 1) / PTS_PER_BLOCK;
    if (nblocks > 0)
        hashgrid_encode_kernel<<<nblocks, BLOCK_THREADS, 0, stream>>>(x, tables, out, B, rl);
}